// PrototypicalNetwork_69595650064482
// MI455X (gfx1250) — compile-verified
//
#include <hip/hip_runtime.h>
#include <hip/hip_bf16.h>

// ---- problem constants (from reference) ----
#define N_SUPPORT 12800
#define N_QUERY   32768
#define SEQ_LEN   100
#define INPUT_DIM 32
#define EMBED_DIM 64
#define N_CLASSES 128
#define EPS_F     1e-8f

typedef __attribute__((ext_vector_type(2))) float v2f;
typedef __attribute__((ext_vector_type(8))) float v8f;

// -----------------------------------------------------------------------------
// Encode: e = meanpool_seq(x) @ W + b   (one wave32 per item)
// Streaming-bound: 583MB total input -> this kernel pair is the runtime floor.
// -----------------------------------------------------------------------------
__global__ void proto_encode_kernel(const float* __restrict__ x,
                                    const float* __restrict__ W,
                                    const float* __restrict__ bias,
                                    float* __restrict__ e,
                                    float* __restrict__ norm2,  // null for support
                                    int n_items) {
    const int lane = threadIdx.x & 31;
    const int item = blockIdx.x * (blockDim.x >> 5) + (threadIdx.x >> 5);
    if (item >= n_items) return;

    // mean over SEQ_LEN for this lane's input dim (coalesced 128B per timestep)
    const float* xi = x + (size_t)item * (SEQ_LEN * INPUT_DIM) + lane;
    float s = 0.0f;
#pragma unroll 5
    for (int t = 0; t < SEQ_LEN; ++t)
        s += xi[(size_t)t * INPUT_DIM];
    const float mean = s * (1.0f / (float)SEQ_LEN);

    // projection: each lane produces output dims (2*lane, 2*lane+1)
    float a0 = 0.0f, a1 = 0.0f;
#pragma unroll
    for (int d = 0; d < INPUT_DIM; ++d) {
        const float md = __shfl(mean, d, 32);
        a0 = fmaf(md, W[d * EMBED_DIM + 2 * lane + 0], a0);
        a1 = fmaf(md, W[d * EMBED_DIM + 2 * lane + 1], a1);
    }
    a0 += bias[2 * lane + 0];
    a1 += bias[2 * lane + 1];

    float* eo = e + (size_t)item * EMBED_DIM;
    eo[2 * lane + 0] = a0;
    eo[2 * lane + 1] = a1;

    if (norm2 != nullptr) {
        float sq = a0 * a0 + a1 * a1;
#pragma unroll
        for (int off = 16; off > 0; off >>= 1)
            sq += __shfl_xor(sq, off, 32);
        if (lane == 0) norm2[item] = sq;
    }
}

// -----------------------------------------------------------------------------
// Prototypes: per-class mean of support embeddings (deterministic segmented
// reduction; one block per class, 4 item-slices x 64 dims = 256 threads).
// Also emits ||p||^2 per class.
// -----------------------------------------------------------------------------
__global__ void proto_proto_kernel(const float* __restrict__ se,
                                   const int* __restrict__ labels,
                                   float* __restrict__ proto_out,  // d_out tail
                                   float* __restrict__ proto_ws,
                                   float* __restrict__ pnorm2) {
    const int c = blockIdx.x;        // class
    const int t = threadIdx.x;       // 0..255
    const int j = t & 63;            // embed dim
    const int s = t >> 6;            // slice 0..3
    const int per = N_SUPPORT / 4;   // 3200

    float sum = 0.0f;
    int cnt = 0;
    for (int k = 0; k < per; ++k) {
        const int i = s * per + k;
        if (labels[i] == c) {        // uniform per wave -> scalar branch
            sum += se[(size_t)i * EMBED_DIM + j];
            ++cnt;
        }
    }

    __shared__ float ps[4][EMBED_DIM];
    __shared__ int   pc[4];
    __shared__ float sq[EMBED_DIM];
    ps[s][j] = sum;
    if (j == 0) pc[s] = cnt;
    __syncthreads();

    if (t < EMBED_DIM) {
        const float tot = ps[0][j] + ps[1][j] + ps[2][j] + ps[3][j];
        const int   n   = pc[0] + pc[1] + pc[2] + pc[3];
        const float p   = tot / (float)n;
        proto_out[c * EMBED_DIM + j] = p;
        proto_ws[c * EMBED_DIM + j]  = p;
        sq[j] = p * p;
    }
    __syncthreads();

    if (t == 0) {
        float nn = 0.0f;
#pragma unroll
        for (int jj = 0; jj < EMBED_DIM; ++jj) nn += sq[jj];
        pnorm2[c] = nn;
    }
}

// -----------------------------------------------------------------------------
// Distances via WMMA:  logits[q,c] = -( ||q||^2 + ||p||^2 - 2 q.p + EPS )
// q.p computed with V_WMMA_F32_16X16X4_F32 tiles (K=64 -> 16 wmma per tile).
// Block = 256 threads (8 waves); block covers 128 queries x all 128 classes.
// Prototypes (32KB) staged in LDS.
// -----------------------------------------------------------------------------
__global__ void proto_dist_kernel(const float* __restrict__ qe,
                                  const float* __restrict__ qnorm2,
                                  const float* __restrict__ proto,
                                  const float* __restrict__ pnorm2,
                                  float* __restrict__ logits) {
    __shared__ float Plds[N_CLASSES * EMBED_DIM];  // 32 KB
    __shared__ float pn2[N_CLASSES];
    __shared__ float qn2[128];

    const int tid   = threadIdx.x;
    const int qbase = blockIdx.x * 128;

    for (int idx = tid; idx < N_CLASSES * EMBED_DIM; idx += 256)
        Plds[idx] = proto[idx];
    if (tid < N_CLASSES) {
        pn2[tid] = pnorm2[tid];
        qn2[tid] = qnorm2[qbase + tid];
    }
    __syncthreads();

    const int w    = tid >> 5;        // wave id 0..7 -> 16-query stripe
    const int l    = tid & 31;
    const int half = l >> 4;          // lane half selects K pair (ISA A/B layout)
    const int r    = l & 15;          // row (M for A) / col (N for B)
    const int q0   = qbase + w * 16;

    // Preload A fragments: A[m=r, k=4s+2*half+{0,1}]
    v2f afrag[16];
    const float* qrow = qe + (size_t)(q0 + r) * EMBED_DIM + 2 * half;
#pragma unroll
    for (int s = 0; s < 16; ++s)
        afrag[s] = *(const v2f*)(qrow + 4 * s);

#pragma unroll
    for (int nt = 0; nt < 8; ++nt) {
        v8f acc = {};
        // B fragments from LDS: B[k=4s+2*half+{0,1}, n=r] == P[n][k]
        const float* prow = &Plds[(nt * 16 + r) * EMBED_DIM + 2 * half];
#pragma unroll
        for (int s = 0; s < 16; ++s) {
            v2f bfrag = *(const v2f*)(prow + 4 * s);
            acc = __builtin_amdgcn_wmma_f32_16x16x4_f32(
                /*neg_a=*/false, afrag[s],
                /*neg_b=*/false, bfrag,
                /*c_mod=*/(short)0, acc,
                /*reuse_a=*/false, /*reuse_b=*/false);
        }
        // Epilogue: C/D layout -> lane holds col n=r, rows m = v + 8*half
        const int   n  = nt * 16 + r;
        const float pn = pn2[n];
#pragma unroll
        for (int v = 0; v < 8; ++v) {
            const int m = v + 8 * half;
            const float d2 = qn2[w * 16 + m] + pn - 2.0f * acc[v] + EPS_F;
            logits[(size_t)(q0 + m) * N_CLASSES + n] = -d2;
        }
    }
}

// -----------------------------------------------------------------------------
extern "C" void kernel_launch(void* const* d_in, const int* in_sizes, int n_in,
                              void* d_out, int out_size, void* d_ws, size_t ws_size,
                              hipStream_t stream) {
    const float* support = (const float*)d_in[0];
    const int*   labels  = (const int*)d_in[1];
    const float* query   = (const float*)d_in[2];
    const float* W       = (const float*)d_in[3];
    const float* bias    = (const float*)d_in[4];

    float* out       = (float*)d_out;
    float* logits    = out;                                   // 32768*128
    float* proto_out = out + (size_t)N_QUERY * N_CLASSES;     // 128*64

    // workspace layout (floats)
    float* ws   = (float*)d_ws;
    float* se   = ws;                                         // 12800*64
    float* qe   = se + (size_t)N_SUPPORT * EMBED_DIM;         // 32768*64
    float* qn2  = qe + (size_t)N_QUERY * EMBED_DIM;           // 32768
    float* pn2  = qn2 + N_QUERY;                              // 128
    float* pws  = pn2 + N_CLASSES;                            // 128*64

    // encode: 8 waves (items) per 256-thread block
    proto_encode_kernel<<<N_SUPPORT / 8, 256, 0, stream>>>(support, W, bias, se, nullptr, N_SUPPORT);
    proto_encode_kernel<<<N_QUERY / 8, 256, 0, stream>>>(query, W, bias, qe, qn2, N_QUERY);

    // prototypes (one block per class)
    proto_proto_kernel<<<N_CLASSES, 256, 0, stream>>>(se, labels, proto_out, pws, pn2);

    // distances (256 blocks x 128 queries)
    proto_dist_kernel<<<N_QUERY / 128, 256, 0, stream>>>(qe, qn2, pws, pn2, logits);
}